// GeoGNNBlock_5111011083034
// MI455X (gfx1250) — compile-verified
//
#include <hip/hip_runtime.h>

#define D 128           // embed dim
#define H 256           // hidden dim (2*D)
#define GMAX 512        // graphs per batch (reference G)
#define LN_EPS 1e-5f

typedef float v2f __attribute__((ext_vector_type(2)));
typedef float v8f __attribute__((ext_vector_type(8)));

// ---------------------------------------------------------------------------
// Zero-fill scratch (agg accumulator + graph counts) — must run every launch.
// ---------------------------------------------------------------------------
__global__ void zero_kernel(float* __restrict__ p, long n) {
  long i = (long)blockIdx.x * blockDim.x + threadIdx.x;
  const long stride = (long)gridDim.x * blockDim.x;
  for (; i < n; i += stride) p[i] = 0.0f;
}

// ---------------------------------------------------------------------------
// Repack a [K][C] row-major weight matrix into WMMA-B-fragment pairs:
// dst[(k/2)*C + c] = float2{ src[k][c], src[k+1][c] }  (k even).
// Each B-fragment in the GEMM then needs ONE global_load_b64 instead of two
// b32 loads from rows 1KB apart. 256KB total, trivial vs the edge phase.
// ---------------------------------------------------------------------------
__global__ void pack_pairs_kernel(const float* __restrict__ src,
                                  float* __restrict__ dst, int K, int C) {
  const int id = blockIdx.x * blockDim.x + threadIdx.x;
  const int total = (K / 2) * C;
  if (id >= total) return;
  const int kr2 = id / C;
  const int col = id - kr2 * C;
  v2f p;
  p.x = src[(size_t)(2 * kr2) * C + col];
  p.y = src[(size_t)(2 * kr2 + 1) * C + col];
  *(v2f*)&dst[(size_t)id * 2] = p;
}

// ---------------------------------------------------------------------------
// Phase 1: per-edge message relu(x[src] + e) scatter-added into agg[dst].
// One wave32 per edge, each lane owns one float4 chunk of the 128-dim row.
// Atomic adds land in L2 (agg is 25.6MB << 192MB L2).
// ---------------------------------------------------------------------------
__global__ void __launch_bounds__(256)
edge_msg_kernel(const float* __restrict__ xh, const float* __restrict__ eh,
                const int* __restrict__ src, const int* __restrict__ dst,
                float* __restrict__ agg, int E) {
  const long gid = (long)blockIdx.x * blockDim.x + threadIdx.x;
  const int e = (int)(gid >> 5);
  if (e >= E) return;
  const int c = ((int)gid & 31) * 4;
  const int s = src[e];
  const int d = dst[e];
  const float4 x = *(const float4*)(xh + (size_t)s * D + c);
  const float4 ev = *(const float4*)(eh + (size_t)e * D + c);
  float4 m;
  m.x = fmaxf(x.x + ev.x, 0.0f);
  m.y = fmaxf(x.y + ev.y, 0.0f);
  m.z = fmaxf(x.z + ev.z, 0.0f);
  m.w = fmaxf(x.w + ev.w, 0.0f);
  float* ap = agg + (size_t)d * D + c;
  atomicAdd(ap + 0, m.x);
  atomicAdd(ap + 1, m.y);
  atomicAdd(ap + 2, m.z);
  atomicAdd(ap + 3, m.w);
}

// ---------------------------------------------------------------------------
// GraphNorm counts: histogram of node_id (values < GMAX).
// ---------------------------------------------------------------------------
__global__ void graph_count_kernel(const int* __restrict__ nid,
                                   float* __restrict__ counts, int n) {
  const int i = blockIdx.x * blockDim.x + threadIdx.x;
  if (i < n) atomicAdd(&counts[nid[i]], 1.0f);
}

// ---------------------------------------------------------------------------
// Phase 2+3 fused: h = x + agg; t = relu(h@W1 + b1); y = t@W2 + b2;
// LayerNorm(y) * gamma + beta; GraphNorm scale; ReLU; + x.
//
// One wave32 owns 16 rows. WMMA f32 16x16x4:
//   A 16x4 layout : lanes 0-15 hold M=lane {K=0,K=1}; lanes 16-31 {K=2,K=3}
//   B 4x16 layout : v0 = rows K=0 (lanes 0-15) / K=2 (lanes 16-31); v1 = K=1/K=3
//   C/D 16x16     : vgpr r = row r (lanes 0-15) / row r+8 (lanes 16-31)
// Stage-1 tiles are bounced through a wave-private LDS staging tile to
// convert C-layout -> A-layout for stage 2 (intermediate [N,256] never
// leaves the WGP). W1p/W2p are the pair-packed weights (one b64 per B-frag).
// ---------------------------------------------------------------------------
__global__ void __launch_bounds__(256)
gine_mlp_ln_kernel(const float* __restrict__ xh, const float* __restrict__ agg,
                   const float* __restrict__ W1p, const float* __restrict__ b1,
                   const float* __restrict__ W2p, const float* __restrict__ b2,
                   const float* __restrict__ gamma, const float* __restrict__ beta,
                   const int* __restrict__ nid, const float* __restrict__ counts,
                   float* __restrict__ out, int N) {
  __shared__ float hs[8][16][132];  // per-wave h tile, padded vs bank conflicts
  __shared__ float st[8][16][20];   // per-wave stage-1 staging tile

  const int wv = threadIdx.x >> 5;
  const int lane = threadIdx.x & 31;
  const int half = lane >> 4;       // 0: lanes 0-15, 1: lanes 16-31
  const int l15 = lane & 15;
  const int r0 = (blockIdx.x * 8 + wv) * 16;

  // --- load h = x + agg for my 16 rows (coalesced: one row per iteration) ---
  for (int i = 0; i < 16; ++i) {
    int row = r0 + i;
    if (row >= N) row = N - 1;  // clamp tail; stores are guarded later
    const float4 a = *(const float4*)(xh + (size_t)row * D + lane * 4);
    const float4 g = *(const float4*)(agg + (size_t)row * D + lane * 4);
    float4 s;
    s.x = a.x + g.x; s.y = a.y + g.y; s.z = a.z + g.z; s.w = a.w + g.w;
    *(float4*)&hs[wv][i][lane * 4] = s;
  }
  asm volatile("s_wait_dscnt 0" ::: "memory");  // wave-private tile: no barrier

  v8f o[8];  // final 16x128 accumulators (8 N-tiles of 16)
  {
    const v8f z = {0.f, 0.f, 0.f, 0.f, 0.f, 0.f, 0.f, 0.f};
#pragma unroll
    for (int m = 0; m < 8; ++m) o[m] = z;
  }

  for (int n = 0; n < 16; ++n) {  // stage-1 N-tiles over H=256
    v8f acc = {0.f, 0.f, 0.f, 0.f, 0.f, 0.f, 0.f, 0.f};
#pragma unroll 4
    for (int j = 0; j < 32; ++j) {  // K over D=128, 4 per step
      const v2f a = *(const v2f*)&hs[wv][l15][4 * j + 2 * half];
      // packed pair (rows 4j+2h, 4j+2h+1) at column n*16+l15: one b64 load
      const v2f b =
          *(const v2f*)&W1p[((size_t)(2 * j + half) * H + n * 16 + l15) * 2];
      acc = __builtin_amdgcn_wmma_f32_16x16x4_f32(
          false, a, false, b, (short)0, acc, false, false);
    }
    // bias + relu, stash tile in A-layout-readable LDS
    const float bias = b1[n * 16 + l15];
#pragma unroll
    for (int i = 0; i < 8; ++i) {
      const float v = acc[i] + bias;
      st[wv][i + 8 * half][l15] = v > 0.f ? v : 0.f;
    }
    asm volatile("s_wait_dscnt 0" ::: "memory");
    // stage 2: this tile covers K range [16n, 16n+16) of the 256-K GEMM
#pragma unroll
    for (int kk = 0; kk < 4; ++kk) {
      const v2f a2 = *(const v2f*)&st[wv][l15][4 * kk + 2 * half];
      const int kp2 = n * 8 + 2 * kk + half;  // pair index into packed W2
#pragma unroll
      for (int m = 0; m < 8; ++m) {
        const v2f b = *(const v2f*)&W2p[((size_t)kp2 * D + m * 16 + l15) * 2];
        o[m] = __builtin_amdgcn_wmma_f32_16x16x4_f32(
            false, a2, false, b, (short)0, o[m], false, false);
      }
    }
    asm volatile("s_wait_dscnt 0" ::: "memory");  // staging WAR vs next n
  }

  // --- bias2 + LayerNorm + GraphNorm + ReLU + residual, all in C-layout ---
  float gm[8], bt[8];
#pragma unroll
  for (int m = 0; m < 8; ++m) {
    const int col = m * 16 + l15;
    const float bias2 = b2[col];
    gm[m] = gamma[col];
    bt[m] = beta[col];
#pragma unroll
    for (int i = 0; i < 8; ++i) o[m][i] += bias2;
  }

  // row stats: each 16-lane half holds one row per accumulator index i
  float mu[8], rs[8];
#pragma unroll
  for (int i = 0; i < 8; ++i) {
    float s = 0.f, q = 0.f;
#pragma unroll
    for (int m = 0; m < 8; ++m) {
      const float v = o[m][i];
      s += v;
      q += v * v;
    }
#pragma unroll
    for (int d = 1; d < 16; d <<= 1) {  // stays within the 16-lane half
      s += __shfl_xor(s, d, 32);
      q += __shfl_xor(q, d, 32);
    }
    const float mean = s * (1.0f / 128.0f);
    float var = q * (1.0f / 128.0f) - mean * mean;
    var = var < 0.f ? 0.f : var;
    mu[i] = mean;
    rs[i] = rsqrtf(var + LN_EPS);
  }

#pragma unroll
  for (int i = 0; i < 8; ++i) {
    const int row = r0 + i + 8 * half;
    const int rc = row < N ? row : N - 1;
    const float cnt = counts[nid[rc]];
    const float gscale = rsqrtf(cnt < 1.f ? 1.f : cnt);
    if (row < N) {
#pragma unroll
      for (int m = 0; m < 8; ++m) {
        const int col = m * 16 + l15;
        float v = (o[m][i] - mu[i]) * rs[i] * gm[m] + bt[m];
        v *= gscale;
        v = v > 0.f ? v : 0.f;
        out[(size_t)row * D + col] = v + xh[(size_t)row * D + col];
      }
    }
  }
}

// ---------------------------------------------------------------------------
extern "C" void kernel_launch(void* const* d_in, const int* in_sizes, int n_in,
                              void* d_out, int out_size, void* d_ws, size_t ws_size,
                              hipStream_t stream) {
  const float* xh    = (const float*)d_in[0];  // node_hidden [N,D]
  const float* eh    = (const float*)d_in[1];  // edge_hidden [E,D]
  const float* W1    = (const float*)d_in[2];  // [D,2D]
  const float* b1    = (const float*)d_in[3];  // [2D]
  const float* W2    = (const float*)d_in[4];  // [2D,D]
  const float* b2    = (const float*)d_in[5];  // [D]
  const float* gamma = (const float*)d_in[6];  // [D]
  const float* beta  = (const float*)d_in[7];  // [D]
  const int*   ei    = (const int*)d_in[8];    // [2,E] (src row, dst row)
  const int*   nid   = (const int*)d_in[9];    // [N]
  // d_in[10] = edge_id (unused by the reference math)
  float* out = (float*)d_out;

  const int N = in_sizes[0] / D;
  const int E = in_sizes[1] / D;

  // ws layout: counts[GMAX] | W1p[D*H] | W2p[H*D] | agg[N*D]
  // (agg falls back to d_out if ws is too small; safe because each block
  // reads its agg rows before writing its out rows, and rows are private.)
  float* counts = (float*)d_ws;
  float* W1p = counts + GMAX;
  float* W2p = W1p + (size_t)D * H;
  float* aggws = W2p + (size_t)H * D;
  const size_t need = ((size_t)GMAX + 2 * (size_t)D * H + (size_t)N * D) * sizeof(float);
  float* agg = (ws_size >= need) ? aggws : out;

  zero_kernel<<<2, 256, 0, stream>>>(counts, (long)GMAX);
  zero_kernel<<<2048, 256, 0, stream>>>(agg, (long)N * D);

  // repack weights into B-fragment pair layout (one b64 per fragment)
  pack_pairs_kernel<<<(D / 2 * H + 255) / 256, 256, 0, stream>>>(W1, W1p, D, H);
  pack_pairs_kernel<<<(H / 2 * D + 255) / 256, 256, 0, stream>>>(W2, W2p, H, D);

  {
    const long threads = (long)E * 32;
    const int blocks = (int)((threads + 255) / 256);
    edge_msg_kernel<<<blocks, 256, 0, stream>>>(xh, eh, ei, ei + E, agg, E);
  }
  graph_count_kernel<<<(N + 255) / 256, 256, 0, stream>>>(nid, counts, N);

  gine_mlp_ln_kernel<<<(N + 127) / 128, 256, 0, stream>>>(
      xh, agg, W1p, b1, W2p, b2, gamma, beta, nid, counts, out, N);
}